// PCWA_45226005627011
// MI455X (gfx1250) — compile-verified
//
#include <hip/hip_runtime.h>
#include <hip/hip_bf16.h>

typedef __attribute__((ext_vector_type(2))) float v2f;
typedef __attribute__((ext_vector_type(8))) float v8f;

// Problem constants (from reference: B=8, C=32, H=W=512, PS=2)
static constexpr int Bn = 8;
static constexpr int Cn = 32;
static constexpr int Hn = 512;
static constexpr int Wn = 512;
// Tile: 2 rows x 128 cols = 256 pixels per workgroup, 256 threads = 8 waves.
static constexpr int TCOLS = 128;
static constexpr int NPIX  = 256;
static constexpr int NBLK  = Bn * (Hn / 2) * (Wn / TCOLS); // 8192

static constexpr int PADN  = NPIX + 8; // 264: 32ch x 256px LDS tile row stride
static constexpr int WSTA  = 68;       // conv1x1 weight LDS row stride

// ---------------------------------------------------------------------------
// Async Global<->LDS DMA helpers (CDNA5 ASYNCcnt path) with safe fallback.
// Probe result: builtins take (AS(1) v4i* gaddr, AS(3) v4i* lds, Ii off, Ii cpol).
// ---------------------------------------------------------------------------
#if defined(__HIP_DEVICE_COMPILE__) && defined(__has_builtin)
#if __has_builtin(__builtin_amdgcn_global_load_async_to_lds_b128) && \
    __has_builtin(__builtin_amdgcn_global_store_async_from_lds_b128)
#define USE_ASYNC_LDS 1
#endif
#endif
#ifndef USE_ASYNC_LDS
#define USE_ASYNC_LDS 0
#endif

#if USE_ASYNC_LDS
typedef int v4i_raw __attribute__((vector_size(16)));
typedef __attribute__((address_space(1))) v4i_raw* gptr4;
typedef __attribute__((address_space(3))) v4i_raw* lptr4;
#endif

__device__ __forceinline__ void g2l_b128(const float* g, float* l) {
#if USE_ASYNC_LDS
  __builtin_amdgcn_global_load_async_to_lds_b128((gptr4)g, (lptr4)l, 0, 0);
#else
  *(float4*)l = *(const float4*)g;
#endif
}

__device__ __forceinline__ void l2g_b128(float* g, const float* l) {
#if USE_ASYNC_LDS
  __builtin_amdgcn_global_store_async_from_lds_b128((gptr4)g, (lptr4)l, 0, 0);
#else
  *(float4*)g = *(const float4*)l;
#endif
}

__device__ __forceinline__ void wait_async0() {
#if USE_ASYNC_LDS
#if __has_builtin(__builtin_amdgcn_s_wait_asynccnt)
  __builtin_amdgcn_s_wait_asynccnt(0);
#else
  asm volatile("s_wait_asynccnt 0x0" ::: "memory");
#endif
#endif
}

// ---------------------------------------------------------------------------
// Kernel A: patch gate + 1x1 conv (GEMM 32x64 x 64xN via v_wmma_f32_16x16x4)
//           + channel max / channel mean.  fs -> d_out (coalesced via LDS),
//           (cm,ca) -> d_ws.
// ---------------------------------------------------------------------------
__launch_bounds__(256)
__global__ void pcwa_gate_conv1x1(const float* __restrict__ x,
                                  const float* __restrict__ mlp_w,
                                  const float* __restrict__ mlp_b,
                                  const float* __restrict__ conv_w,
                                  const float* __restrict__ conv_b,
                                  float* __restrict__ fs,
                                  float* __restrict__ cmca) {
  __shared__ float s_x[Cn * PADN];   // x tile; reused as fs staging in epilogue
  __shared__ float s_g[Cn * 64];     // gate: 32 ch x 64 patches
  __shared__ float s_w[Cn * WSTA];   // conv1x1 weight 32x64 (padded)
  __shared__ float s_bias[Cn];

  const int tid = threadIdx.x;
  const int bid = blockIdx.x;
  const int wt  = Wn / TCOLS;             // 4
  const int rt  = Hn / 2;                 // 256
  const int b   = bid / (wt * rt);
  const int rem = bid % (wt * rt);
  const int h0  = (rem / wt) * 2;
  const int w0  = (rem % wt) * TCOLS;

  const float gw  = mlp_w[0];
  const float gb2 = 2.0f * mlp_b[0];

  // --- stage conv1x1 weights (32x64) + bias ---
  for (int i = tid; i < Cn * 64; i += 256)
    s_w[(i >> 6) * WSTA + (i & 63)] = conv_w[i];
  if (tid < Cn) s_bias[tid] = conv_b[tid];

  // --- load x tile; compute 2x2 patch gate thread-locally ---
  const int cld = tid >> 3;   // channel 0..31
  const int sld = tid & 7;    // 16-col segment 0..7
  {
    const float* xp = x + (((b * Cn + cld) * Hn + h0) * Wn + w0 + 16 * sld);
    float r0[16], r1[16];
#pragma unroll
    for (int j = 0; j < 4; ++j) {
      float4 t0 = ((const float4*)xp)[j];
      float4 t1 = ((const float4*)(xp + Wn))[j];
      r0[4 * j + 0] = t0.x; r0[4 * j + 1] = t0.y;
      r0[4 * j + 2] = t0.z; r0[4 * j + 3] = t0.w;
      r1[4 * j + 0] = t1.x; r1[4 * j + 1] = t1.y;
      r1[4 * j + 2] = t1.z; r1[4 * j + 3] = t1.w;
    }
    float* xr0 = &s_x[cld * PADN + 16 * sld];    // row 0 pixels
    float* xr1 = xr0 + TCOLS;                    // row 1 pixels
#pragma unroll
    for (int j = 0; j < 4; ++j) {
      ((float4*)xr0)[j] = make_float4(r0[4 * j], r0[4 * j + 1], r0[4 * j + 2], r0[4 * j + 3]);
      ((float4*)xr1)[j] = make_float4(r1[4 * j], r1[4 * j + 1], r1[4 * j + 2], r1[4 * j + 3]);
    }
#pragma unroll
    for (int p = 0; p < 8; ++p) {  // 8 full 2x2 patches per thread
      const float e0 = r0[2 * p], e1 = r0[2 * p + 1];
      const float e2 = r1[2 * p], e3 = r1[2 * p + 1];
      const float m  = fmaxf(fmaxf(e0, e1), fmaxf(e2, e3));
      const float a  = 0.25f * (e0 + e1 + e2 + e3);
      const float g  = 1.0f / (1.0f + __expf(-((m + a) * gw + gb2)));
      s_g[cld * 64 + 8 * sld + p] = g;
    }
  }
  __syncthreads();

  // --- GEMM: out[32, 256] = W[32x64] x cat[64, 256]; wave handles 32 pixels ---
  const int lane = tid & 31;
  const int wv   = tid >> 5;
  const int half = lane >> 4;
  const int l15  = lane & 15;
  const int n0   = wv * 32;
  const int nA   = n0 + l15;
  const int nB   = n0 + 16 + l15;
  const int pA   = (nA & 127) >> 1;
  const int pB   = (nB & 127) >> 1;

  const v8f zero = {0.f, 0.f, 0.f, 0.f, 0.f, 0.f, 0.f, 0.f};
  v8f acc[2][2];  // [m-block][n-tile]
  acc[0][0] = zero; acc[0][1] = zero; acc[1][0] = zero; acc[1][1] = zero;

#pragma unroll
  for (int kk = 0; kk < 16; ++kk) {
    const int kb = 4 * kk + 2 * half;  // this lane's K pair: (kb, kb+1)
    const v2f A0 = *(const v2f*)&s_w[l15 * WSTA + kb];
    const v2f A1 = *(const v2f*)&s_w[(16 + l15) * WSTA + kb];
    v2f B0, B1;
    if (kk < 8) {  // cat rows 0..31 = product = x * gate
      B0.x = s_x[kb * PADN + nA]       * s_g[kb * 64 + pA];
      B0.y = s_x[(kb + 1) * PADN + nA] * s_g[(kb + 1) * 64 + pA];
      B1.x = s_x[kb * PADN + nB]       * s_g[kb * 64 + pB];
      B1.y = s_x[(kb + 1) * PADN + nB] * s_g[(kb + 1) * 64 + pB];
    } else {       // cat rows 32..63 = x
      const int kc = kb - 32;
      B0.x = s_x[kc * PADN + nA];
      B0.y = s_x[(kc + 1) * PADN + nA];
      B1.x = s_x[kc * PADN + nB];
      B1.y = s_x[(kc + 1) * PADN + nB];
    }
    acc[0][0] = __builtin_amdgcn_wmma_f32_16x16x4_f32(false, A0, false, B0, (short)0, acc[0][0], false, false);
    acc[1][0] = __builtin_amdgcn_wmma_f32_16x16x4_f32(false, A1, false, B0, (short)0, acc[1][0], false, false);
    acc[0][1] = __builtin_amdgcn_wmma_f32_16x16x4_f32(false, A0, false, B1, (short)0, acc[0][1], false, false);
    acc[1][1] = __builtin_amdgcn_wmma_f32_16x16x4_f32(false, A1, false, B1, (short)0, acc[1][1], false, false);
  }

  // --- epilogue: +bias; channel max/mean -> cmca; stage fs tile into s_x.
  //     A wave writes only its own 32 pixel-columns of s_x (rows 0..31),
  //     which no other wave reads during its GEMM -> no barrier needed here.
#pragma unroll
  for (int t = 0; t < 2; ++t) {
    const int n  = n0 + 16 * t + l15;
    const int hh = h0 + (n >> 7);
    const int ww = w0 + (n & 127);
    float mx = -3.402823466e38f, sm = 0.f;
#pragma unroll
    for (int m = 0; m < 2; ++m) {
#pragma unroll
      for (int v = 0; v < 8; ++v) {
        const int ch = 16 * m + 8 * half + v;   // D layout: VGPR v, half -> M+8
        const float val = acc[m][t][v] + s_bias[ch];
        s_x[ch * PADN + n] = val;
        mx = fmaxf(mx, val);
        sm += val;
      }
    }
    mx = fmaxf(mx, __shfl_xor(mx, 16, 32));
    sm = sm + __shfl_xor(sm, 16, 32);
    if (half == 0) {
      cmca[((b * 2 + 0) * Hn + hh) * Wn + ww] = mx;
      cmca[((b * 2 + 1) * Hn + hh) * Wn + ww] = sm * (1.0f / 32.0f);
    }
  }
  __syncthreads();

  // --- cooperative fully-coalesced fs store (LDS -> global, async DMA) ---
  {
    float* fp = fs + (((b * Cn + cld) * Hn + h0) * Wn + w0 + 16 * sld);
    const float* l0 = &s_x[cld * PADN + 16 * sld];
#pragma unroll
    for (int j = 0; j < 4; ++j) {
      l2g_b128(fp + 4 * j,      l0 + 4 * j);
      l2g_b128(fp + Wn + 4 * j, l0 + TCOLS + 4 * j);
    }
  }
  // S_ENDPGM performs an implicit wait-idle, covering outstanding async stores.
}

// ---------------------------------------------------------------------------
// Kernel B: 7x7 conv on (cm,ca) as im2col GEMM (32 x 98pad100 x N) via WMMA;
//           fs tile async-prefetched into LDS at kernel start (overlaps GEMM);
//           out = sigmoid(att + b) * fs, written back coalesced from LDS.
// ---------------------------------------------------------------------------
static constexpr int HST  = 136;        // halo row stride (134 used)
static constexpr int HCST = 8 * HST;    // halo per-channel stride
static constexpr int WSTB = 104;        // 7x7 weight LDS row stride (K=100 used)

__device__ __forceinline__ int koff(int k) {  // folds to constants after unroll
  if (k >= 98) return 0;                      // zero-padded K rows (A rows are 0)
  const int ic = k / 49;
  const int r  = k % 49;
  return ic * HCST + (r / 7) * HST + (r % 7);
}

__launch_bounds__(256)
__global__ void pcwa_conv7_gate(const float* __restrict__ cmca,
                                const float* __restrict__ convout_w,
                                const float* __restrict__ convout_b,
                                float* __restrict__ out) {
  __shared__ float s_halo[2 * HCST];    // 2 ch x 8 rows x 136
  __shared__ float s_wc[Cn * WSTB];     // 32 x 100 (98 real + 2 zero)
  __shared__ float s_cb[Cn];
  __shared__ float s_fs[Cn * PADN];     // fs tile staging (32 ch x 256 px)

  const int tid = threadIdx.x;
  const int bid = blockIdx.x;
  const int wt  = Wn / TCOLS;
  const int rt  = Hn / 2;
  const int b   = bid / (wt * rt);
  const int rem = bid % (wt * rt);
  const int h0  = (rem / wt) * 2;
  const int w0  = (rem % wt) * TCOLS;

  const int cld = tid >> 3;
  const int sld = tid & 7;

  // --- async-prefetch fs tile into LDS; completes under the GEMM ---
  {
    const float* fp = out + (((b * Cn + cld) * Hn + h0) * Wn + w0 + 16 * sld);
    float* l0 = &s_fs[cld * PADN + 16 * sld];
#pragma unroll
    for (int j = 0; j < 4; ++j) {
      g2l_b128(fp + 4 * j,      l0 + 4 * j);
      g2l_b128(fp + Wn + 4 * j, l0 + TCOLS + 4 * j);
    }
  }

  // --- halo load (8 rows x 134 cols x 2 ch), zero outside the image ---
  for (int i = tid; i < 2 * 8 * 134; i += 256) {
    const int ic = i / (8 * 134);
    const int r  = (i % (8 * 134)) / 134;
    const int cc = i % 134;
    const int gh = h0 - 3 + r;
    const int gwc = w0 - 3 + cc;
    float v = 0.f;
    if ((unsigned)gh < (unsigned)Hn && (unsigned)gwc < (unsigned)Wn)
      v = cmca[((b * 2 + ic) * Hn + gh) * Wn + gwc];
    s_halo[ic * HCST + r * HST + cc] = v;
  }
  // --- weights 32x98 -> 32x100 (zero pad) ---
  for (int i = tid; i < Cn * 100; i += 256) {
    const int o = i / 100, k = i % 100;
    s_wc[o * WSTB + k] = (k < 98) ? convout_w[o * 98 + k] : 0.f;
  }
  if (tid < Cn) s_cb[tid] = convout_b[tid];
  __syncthreads();

  const int lane = tid & 31;
  const int wv   = tid >> 5;
  const int half = lane >> 4;
  const int l15  = lane & 15;
  const int n0   = wv * 32;
  const int nA   = n0 + l15;
  const int nB   = n0 + 16 + l15;
  const int base0 = (nA >> 7) * HST + (nA & 127);
  const int base1 = (nB >> 7) * HST + (nB & 127);

  const v8f zero = {0.f, 0.f, 0.f, 0.f, 0.f, 0.f, 0.f, 0.f};
  v8f acc[2][2];
  acc[0][0] = zero; acc[0][1] = zero; acc[1][0] = zero; acc[1][1] = zero;

#pragma unroll
  for (int kk = 0; kk < 25; ++kk) {
    const int k0 = 4 * kk;
    const int kb = k0 + 2 * half;
    // im2col offsets: compile-time per kk, runtime select by half
    const int o0 = koff(k0)     + half * (koff(k0 + 2) - koff(k0));
    const int o1 = koff(k0 + 1) + half * (koff(k0 + 3) - koff(k0 + 1));
    const v2f A0 = *(const v2f*)&s_wc[l15 * WSTB + kb];
    const v2f A1 = *(const v2f*)&s_wc[(16 + l15) * WSTB + kb];
    v2f B0, B1;
    B0.x = s_halo[base0 + o0];
    B0.y = s_halo[base0 + o1];
    B1.x = s_halo[base1 + o0];
    B1.y = s_halo[base1 + o1];
    acc[0][0] = __builtin_amdgcn_wmma_f32_16x16x4_f32(false, A0, false, B0, (short)0, acc[0][0], false, false);
    acc[1][0] = __builtin_amdgcn_wmma_f32_16x16x4_f32(false, A1, false, B0, (short)0, acc[1][0], false, false);
    acc[0][1] = __builtin_amdgcn_wmma_f32_16x16x4_f32(false, A0, false, B1, (short)0, acc[0][1], false, false);
    acc[1][1] = __builtin_amdgcn_wmma_f32_16x16x4_f32(false, A1, false, B1, (short)0, acc[1][1], false, false);
  }

  // --- drain fs prefetch: each wave waits its own ASYNCcnt, then barrier ---
  wait_async0();
  __syncthreads();

  // --- epilogue: s_fs <- sigmoid(att + b) * s_fs (in place, own elements) ---
#pragma unroll
  for (int t = 0; t < 2; ++t) {
    const int n = n0 + 16 * t + l15;
#pragma unroll
    for (int m = 0; m < 2; ++m) {
#pragma unroll
      for (int v = 0; v < 8; ++v) {
        const int ch  = 16 * m + 8 * half + v;
        const float a = acc[m][t][v] + s_cb[ch];
        const float g = 1.0f / (1.0f + __expf(-a));
        const int li  = ch * PADN + n;
        s_fs[li] = g * s_fs[li];
      }
    }
  }
  __syncthreads();

  // --- cooperative fully-coalesced final store (LDS -> global, async DMA) ---
  {
    float* op = out + (((b * Cn + cld) * Hn + h0) * Wn + w0 + 16 * sld);
    const float* l0 = &s_fs[cld * PADN + 16 * sld];
#pragma unroll
    for (int j = 0; j < 4; ++j) {
      l2g_b128(op + 4 * j,      l0 + 4 * j);
      l2g_b128(op + Wn + 4 * j, l0 + TCOLS + 4 * j);
    }
  }
}

// ---------------------------------------------------------------------------
extern "C" void kernel_launch(void* const* d_in, const int* in_sizes, int n_in,
                              void* d_out, int out_size, void* d_ws, size_t ws_size,
                              hipStream_t stream) {
  (void)in_sizes; (void)n_in; (void)out_size; (void)ws_size;
  const float* x         = (const float*)d_in[0];
  const float* mlp_w     = (const float*)d_in[1];
  const float* mlp_b     = (const float*)d_in[2];
  const float* conv_w    = (const float*)d_in[3];
  const float* conv_b    = (const float*)d_in[4];
  const float* convout_w = (const float*)d_in[5];
  const float* convout_b = (const float*)d_in[6];

  float* fs   = (float*)d_out;  // fs staged in d_out, overwritten by kernel B
  float* cmca = (float*)d_ws;   // (B, 2, H, W) = 16.8 MB scratch

  dim3 grid(NBLK), block(256);
  pcwa_gate_conv1x1<<<grid, block, 0, stream>>>(x, mlp_w, mlp_b, conv_w, conv_b, fs, cmca);
  pcwa_conv7_gate<<<grid, block, 0, stream>>>(cmca, convout_w, convout_b, fs);
}